// DenseAtt_21131239097072
// MI455X (gfx1250) — compile-verified
//
#include <hip/hip_runtime.h>
#include <hip/hip_bf16.h>

typedef float v2f __attribute__((ext_vector_type(2)));
typedef float v4f __attribute__((ext_vector_type(4)));
typedef float v8f __attribute__((ext_vector_type(8)));

// ---------------------------------------------------------------------------
// Phase 1: S = X (16384x128) * [w_left | w_right] (128x2) via
// V_WMMA_F32_16X16X4_F32. One wave (32 lanes) computes 16 rows; K=128 is
// accumulated over 32 WMMA steps of K=4.
//
// A layout (16x4 f32): lane = (Khalf<<4) + M ; reg r holds K = 2*Khalf + r
//   -> per lane a contiguous 8-byte (v2f) load from x[row*128 + k0 + 2*Khalf].
// B layout (4x16 f32): lane = (Khalf<<4) + N ; reg r holds K = 2*Khalf + r
//   -> column 0 = w_left chunk, column 1 = w_right chunk, others zero.
//   B values are loaded UNCONDITIONALLY from a valid address (dead lanes read
//   column-0 data) and zeroed with a branchless select -> v_cndmask, no EXEC
//   branching around loads.
// C/D layout (16x16 f32): lane = (Mhalf<<4) + N ; reg m holds M = 8*Mhalf + m.
// ---------------------------------------------------------------------------
__global__ __launch_bounds__(32) void
denseatt_proj_wmma(const float* __restrict__ x, const float* __restrict__ W,
                   float* __restrict__ sl, float* __restrict__ sr) {
  const int lane  = threadIdx.x;        // 0..31
  const int col   = lane & 15;          // M for A, N for B
  const int khalf = (lane >> 4) << 1;   // 0 or 2
  const int rowBase = blockIdx.x << 4;  // 16 rows per wave

  const float* aptr = x + (size_t)(rowBase + col) * 128 + khalf;
  // Always-valid pointer: live lanes (col<2) point at their W column,
  // dead lanes alias column 0 (result masked to zero below).
  const bool  liveB = (col < 2);
  const float* wptr = W + (liveB ? col : 0) * 128 + khalf;

  v8f c = {0.f, 0.f, 0.f, 0.f, 0.f, 0.f, 0.f, 0.f};
  for (int k0 = 0; k0 < 128; k0 += 4) {
    v2f a  = *(const v2f*)(aptr + k0);   // global_load_b64 (A fragment)
    v2f wv = *(const v2f*)(wptr + k0);   // global_load_b64 (unconditional)
    v2f bf;
    bf.x = liveB ? wv.x : 0.0f;          // v_cndmask_b32, branchless
    bf.y = liveB ? wv.y : 0.0f;
    c = __builtin_amdgcn_wmma_f32_16x16x4_f32(
        /*neg_a=*/false, a, /*neg_b=*/false, bf,
        /*c_mod=*/(short)0, c, /*reuse_a=*/false, /*reuse_b=*/false);
  }

  // Column 0 -> s_left, column 1 -> s_right. Rows M = 8*(lane>>4) + m.
  const int mbase = (lane >> 4) << 3;
  if (col == 0) {
#pragma unroll
    for (int m = 0; m < 8; ++m) sl[rowBase + mbase + m] = c[m];
  } else if (col == 1) {
#pragma unroll
    for (int m = 0; m < 8; ++m) sr[rowBase + mbase + m] = c[m];
  }
}

// ---------------------------------------------------------------------------
// Phase 2: out[b,i,j] = sigmoid(sl[b,i] + sr[b,j] + bias) * adj[b,i,j]
// Pure streaming (268 MB total traffic -> ~11.5 us roofline at 23.3 TB/s):
// one float4 per thread, non-temporal B128 load/store for the single-use
// adj / out streams, regular (cached) loads for the tiny, heavily-reused
// sl/sr vectors.
// ---------------------------------------------------------------------------
__global__ __launch_bounds__(256) void
denseatt_apply(const float* __restrict__ adj, const float* __restrict__ sl,
               const float* __restrict__ sr, const float* __restrict__ bptr,
               float* __restrict__ out) {
  const float bias = bptr[0];
  const int gid = blockIdx.x * 256 + threadIdx.x;   // one float4 per thread
  const int row = gid >> 9;                         // (b*2048 + i), 0..16383
  const int jj  = gid & 511;                        // float4 index in row
  const int b   = row >> 11;

  const float slv = sl[row];
  const v4f  srv  = *(const v4f*)(sr + (b << 11) + (jj << 2));

  const v4f* adjp = (const v4f*)(adj + ((size_t)row << 11)) + jj;
  v4f a4 = __builtin_nontemporal_load(adjp);

  constexpr float L2E = 1.44269504088896340736f;   // log2(e)
  v4f o;
#pragma unroll
  for (int k = 0; k < 4; ++k) {
    float z = slv + srv[k] + bias;
    // sigmoid(z) = 1 / (1 + 2^(-z*log2(e))) with native v_exp_f32 / v_rcp_f32
    float s = __builtin_amdgcn_rcpf(1.0f + __builtin_amdgcn_exp2f(-L2E * z));
    o[k] = a4[k] * s;
  }

  v4f* outp = (v4f*)(out + ((size_t)row << 11)) + jj;
  __builtin_nontemporal_store(o, outp);
}

extern "C" void kernel_launch(void* const* d_in, const int* in_sizes, int n_in,
                              void* d_out, int out_size, void* d_ws, size_t ws_size,
                              hipStream_t stream) {
  const float* x    = (const float*)d_in[0];   // (8,2048,128)
  const float* adj  = (const float*)d_in[1];   // (8,2048,2048)
  const float* W    = (const float*)d_in[2];   // (256,)
  const float* bptr = (const float*)d_in[3];   // scalar
  float* out = (float*)d_out;

  const int R = 8 * 2048;                      // 16384 rows
  float* sl = (float*)d_ws;                    // 16384 floats
  float* sr = sl + R;                          // 16384 floats (128 KB total)

  // Phase 1: 16384 rows / 16 rows-per-wave = 1024 single-wave blocks.
  denseatt_proj_wmma<<<R / 16, 32, 0, stream>>>(x, W, sl, sr);

  // Phase 2: 8*2048*2048/4 float4s / 256 threads = 32768 blocks.
  const int n4 = 8 * 2048 * (2048 / 4);
  denseatt_apply<<<n4 / 256, 256, 0, stream>>>(adj, sl, sr, bptr, out);
}